// ExponentialMovingAverage_55405078118789
// MI455X (gfx1250) — compile-verified
//
#include <hip/hip_runtime.h>
#include <hip/hip_bf16.h>

// EMA scan: h_t = a*x_t + (1-a)*h_{t-1}, a = sigmoid(alpha[d]), over s.
// x: [B=8, S=4096, D=1024] fp32 -> y same shape.
// Bandwidth-bound: chunk-parallel scan + CDNA5 async global->LDS pipeline.

namespace {
constexpr int kB      = 8;
constexpr int kS      = 4096;
constexpr int kD      = 1024;
constexpr int kChunks = 128;            // chunks per sequence
constexpr int kL      = kS / kChunks;   // 32 timesteps per chunk
constexpr int kLog2L  = 5;              // log2(kL)
constexpr int kTS     = 8;              // timesteps per async stage
constexpr int kNS     = kL / kTS;       // 4 stages per chunk
constexpr int kTPB    = kD / 4;         // 256 threads, 4 features each
}

// ---- CDNA5 async global->LDS (inline asm; tracked by ASYNCcnt) -------------
__device__ __forceinline__ void async_load_b128(const float* gaddr, void* lds)
{
    // AS3 offset == low 32 bits of the generic pointer on amdgcn.
    unsigned lds_off      = (unsigned)(size_t)lds;
    unsigned long long ga = (unsigned long long)(size_t)gaddr;
    asm volatile("global_load_async_to_lds_b128 %0, %1, off"
                 :: "v"(lds_off), "v"(ga) : "memory");
}
__device__ __forceinline__ void wait_async_le8() {
    asm volatile("s_wait_asynccnt 8" ::: "memory");
}
__device__ __forceinline__ void wait_async_0() {
    asm volatile("s_wait_asynccnt 0" ::: "memory");
}

__device__ __forceinline__ float4 sigmoid4(float4 v) {
    float4 r;
    r.x = 1.0f / (1.0f + expf(-v.x));
    r.y = 1.0f / (1.0f + expf(-v.y));
    r.z = 1.0f / (1.0f + expf(-v.z));
    r.w = 1.0f / (1.0f + expf(-v.w));
    return r;
}

// ---- Pass 1 (WRITE_Y=false): local scan, emit chunk-final carry ------------
// ---- Pass 3 (WRITE_Y=true):  scan seeded from Hstart, emit y ---------------
// Pipeline order per iteration: wait(stage s) -> compute(stage s) -> issue(s+2).
// The compiler's s_wait_dscnt before the FMAs guarantees the tile reads of
// stage s are complete before the async loads that overwrite buffer (s&1)
// are issued, so the double-buffer WAR hazard is closed architecturally.
template <bool WRITE_Y>
__global__ __launch_bounds__(kTPB)
void ema_scan_kernel(const float* __restrict__ x,
                     const float* __restrict__ alpha,
                     const float* __restrict__ hstart,   // [B*Chunks, D] (pass 3)
                     float*       __restrict__ y,        // (pass 3)
                     float*       __restrict__ carry)    // [B*Chunks, D] (pass 1)
{
    __shared__ __align__(16) float tile[2][kTS][kD];     // 64 KB, double buffered

    const int bc   = blockIdx.x;            // b * kChunks + c
    const int b    = bc / kChunks;
    const int c    = bc % kChunks;
    const int feat = threadIdx.x * 4;

    const float4 a  = sigmoid4(*(const float4*)(alpha + feat));
    float4 om;  om.x = 1.0f - a.x; om.y = 1.0f - a.y; om.z = 1.0f - a.z; om.w = 1.0f - a.w;

    float4 h;
    if (WRITE_Y) h = *(const float4*)(hstart + (size_t)bc * kD + feat);
    else         { h.x = 0.f; h.y = 0.f; h.z = 0.f; h.w = 0.f; }

    const size_t row0  = (size_t)b * kS + (size_t)c * kL;
    const float* xbase = x + row0 * kD + feat;
    float*       ybase = WRITE_Y ? (y + row0 * kD + feat) : nullptr;

    // Prime stages 0 and 1 (16 async-load instrs per wave in flight).
#pragma unroll
    for (int s = 0; s < 2; ++s)
#pragma unroll
        for (int r = 0; r < kTS; ++r)
            async_load_b128(xbase + (size_t)(s * kTS + r) * kD, &tile[s][r][feat]);

#pragma unroll
    for (int s = 0; s < kNS; ++s) {
        // Wait until stage s has landed in LDS.
        if (s == kNS - 1) wait_async_0();
        else              wait_async_le8();

        // Consume stage s.
#pragma unroll
        for (int r = 0; r < kTS; ++r) {
            const float4 xv = *(const float4*)&tile[s & 1][r][feat];
            h.x = fmaf(a.x, xv.x, om.x * h.x);
            h.y = fmaf(a.y, xv.y, om.y * h.y);
            h.z = fmaf(a.z, xv.z, om.z * h.z);
            h.w = fmaf(a.w, xv.w, om.w * h.w);
            if (WRITE_Y)
                *(float4*)(ybase + (size_t)(s * kTS + r) * kD) = h;
        }

        // Refill the buffer we just drained with stage s+2.
        if (s + 2 < kNS) {
#pragma unroll
            for (int r = 0; r < kTS; ++r)
                async_load_b128(xbase + (size_t)((s + 2) * kTS + r) * kD,
                                &tile[s & 1][r][feat]);
        }
    }

    if (!WRITE_Y)
        *(float4*)(carry + (size_t)bc * kD + feat) = h;
}

// ---- Pass 2: sequentially turn carries into true chunk-start states --------
// In-place: carry[b,c] := Hstart[b,c];  Hstart_c = local_{c-1} + dec*Hstart_{c-1}
__global__ __launch_bounds__(256)
void ema_combine_kernel(const float* __restrict__ alpha,
                        float*       __restrict__ carry)
{
    const int idx  = blockIdx.x * 256 + threadIdx.x;   // over B * D/4
    const int b    = idx / (kD / 4);
    const int feat = (idx % (kD / 4)) * 4;

    const float4 a = sigmoid4(*(const float4*)(alpha + feat));
    float4 dec; dec.x = 1.0f - a.x; dec.y = 1.0f - a.y; dec.z = 1.0f - a.z; dec.w = 1.0f - a.w;
#pragma unroll
    for (int i = 0; i < kLog2L; ++i) {                 // (1-a)^kL via squaring
        dec.x *= dec.x; dec.y *= dec.y; dec.z *= dec.z; dec.w *= dec.w;
    }

    float4 H; H.x = 0.f; H.y = 0.f; H.z = 0.f; H.w = 0.f;
    float* base = carry + (size_t)b * kChunks * kD + feat;
    for (int c = 0; c < kChunks; ++c) {
        float4* p = (float4*)(base + (size_t)c * kD);
        const float4 t = *p;           // chunk-local end value
        *p = H;                        // overwrite with chunk-start state
        H.x = fmaf(dec.x, H.x, t.x);
        H.y = fmaf(dec.y, H.y, t.y);
        H.z = fmaf(dec.z, H.z, t.z);
        H.w = fmaf(dec.w, H.w, t.w);
    }
}

extern "C" void kernel_launch(void* const* d_in, const int* in_sizes, int n_in,
                              void* d_out, int out_size, void* d_ws, size_t ws_size,
                              hipStream_t stream)
{
    const float* x     = (const float*)d_in[0];   // [8,4096,1024] fp32
    const float* alpha = (const float*)d_in[1];   // [1024] fp32
    float*       y     = (float*)d_out;
    float*       carry = (float*)d_ws;            // B*Chunks*D floats = 4 MiB

    // Pass 1: per-chunk local scans -> carries
    ema_scan_kernel<false><<<kB * kChunks, kTPB, 0, stream>>>(x, alpha, nullptr, nullptr, carry);
    // Pass 2: carries -> true chunk-start states (in place)
    ema_combine_kernel<<<(kB * kD / 4) / 256, 256, 0, stream>>>(alpha, carry);
    // Pass 3: full scan seeded from chunk-start states -> y
    ema_scan_kernel<true><<<kB * kChunks, kTPB, 0, stream>>>(x, alpha, carry, y, nullptr);
}